// CTCLayer_36301063586438
// MI455X (gfx1250) — compile-verified
//
#include <hip/hip_runtime.h>
#include <math.h>

// CTC forward loss, one workgroup (256 threads / 8 waves) per batch row.
// - y_pred rows staged global->LDS in 16-row (16KB) tiles with
//   global_load_async_to_lds_b128, double-buffered (ASYNCcnt).
// - Per-tile normalizers log(sum_c(p+eps)) computed with V_WMMA_F32_16X16X4_F32
//   (row-sum via multiply-by-ones), off the serial scan's critical path.
// - Serial alpha recursion (lse3 semiring) over T in LDS, 1 barrier/step.

#define NEGV (-1e30f)
#define EPSV (1e-7f)
#define MAXS 256            // supports S = 2L+1 <= 255
#define TROWS 16            // timesteps per tile
#define CDIM 256            // class count (== blockDim.x), blank = CDIM-1
#define ROWSTRIDE 260       // padded LDS row stride (bank-conflict-free column gather)
#define BUFELEMS (TROWS * ROWSTRIDE)

typedef __attribute__((ext_vector_type(2))) float v2f;
typedef __attribute__((ext_vector_type(8))) float v8f;

__global__ __launch_bounds__(256) void ctc_fused_kernel(
    const int*   __restrict__ y_true,   // [B, L]
    const float* __restrict__ y_pred,   // [B, T, C]
    float*       __restrict__ out,      // [B]
    int T, int L)
{
    const int b    = blockIdx.x;
    const int tid  = (int)threadIdx.x;
    const int lane = tid & 31;
    const int wave = tid >> 5;
    const int S    = 2 * L + 1;
    const int blank = CDIM - 1;

    __shared__ __align__(16) float ldsP[2][BUFELEMS];   // double-buffered 16xC tile
    __shared__ float A0[MAXS + 2];                      // alpha ping (front pad = NEG)
    __shared__ float A1[MAXS + 2];                      // alpha pong
    __shared__ int   extcls[MAXS];
    __shared__ unsigned char allowsk[MAXS];
    __shared__ float normAcc[TROWS];
    __shared__ float lognorm[TROWS];

    const float* base  = y_pred + (size_t)b * (size_t)T * (size_t)CDIM; // uniform
    const unsigned pbase = (unsigned)(size_t)&ldsP[0][0];               // LDS byte base
    const int totalElems = T * CDIM;

    // issue async copy of one 16-row tile into buffer (tile & 1)
    auto issue_tile = [&](int tile) {
        const int bufsel = tile & 1;
        const int tbase  = tile * TROWS * CDIM;     // first element of tile
        #pragma unroll
        for (int j = 0; j < 4; ++j) {
            int q    = tid + j * 256;               // float4 chunk id in [0,1024)
            int row  = q >> 6;                      // 64 float4 per row
            int col4 = q & 63;
            int ge   = tbase + q * 4;               // global element index
            if (ge < totalElems) {
                unsigned lb = pbase +
                    (unsigned)(bufsel * BUFELEMS + row * ROWSTRIDE + col4 * 4) * 4u;
                unsigned gb = (unsigned)ge * 4u;
                asm volatile("global_load_async_to_lds_b128 %0, %1, %2"
                             :: "v"(lb), "v"(gb), "s"(base) : "memory");
            }
        }
    };

    // start DMA for tile 0 ASAP
    issue_tile(0);

    // ---- build extended (blank-interleaved) labels + skip mask ----
    if (tid < S) {
        int cls;
        unsigned char allow = 0;
        if (tid & 1) {
            int li = tid >> 1;
            cls = y_true[b * L + li];
            if (tid >= 2) allow = (unsigned char)(cls != y_true[b * L + li - 1]);
        } else {
            cls = blank;
        }
        extcls[tid]  = cls;
        allowsk[tid] = allow;
    }
    if (tid < 2) { A0[tid] = NEGV; A1[tid] = NEGV; }

    float* cur = A0;
    float* nxt = A1;
    const int nTiles = (T + TROWS - 1) / TROWS;

    for (int tile = 0; tile < nTiles; ++tile) {
        if (tile + 1 < nTiles) {
            issue_tile(tile + 1);
            asm volatile("s_wait_asynccnt 4" ::: "memory");  // this tile's 4 done
        } else {
            asm volatile("s_wait_asynccnt 0" ::: "memory");
        }
        __syncthreads();                                     // tile visible to all

        const float* P = &ldsP[tile & 1][0];

        if (tid < TROWS) normAcc[tid] = 0.f;
        __syncthreads();

        // ---- WMMA row sums: D = sum_k A(16x4) * ones(4x16), f32 full precision.
        // Uniform region: EXEC all ones as WMMA requires. Wave w covers K in
        // [32w, 32w+32). A layout: lanes 0-15 hold {K,K+1}, lanes 16-31 {K+2,K+3}.
        {
            const int m     = lane & 15;
            const int khalf = (lane >> 4) << 1;   // 0 or 2
            const int k0    = wave << 5;
            v8f d = {0.f, 0.f, 0.f, 0.f, 0.f, 0.f, 0.f, 0.f};
            v2f ones; ones.x = 1.f; ones.y = 1.f;
            #pragma unroll
            for (int kk = 0; kk < 32; kk += 4) {
                v2f a;
                a.x = P[m * ROWSTRIDE + k0 + kk + khalf];
                a.y = P[m * ROWSTRIDE + k0 + kk + khalf + 1];
                d = __builtin_amdgcn_wmma_f32_16x16x4_f32(
                        false, a, false, ones, (short)0, d, false, false);
            }
            // column N=0 of D lives in lanes 0 (M=i) and 16 (M=8+i)
            if ((lane & 15) == 0) {
                const int mbase = (lane >> 4) << 3;
                #pragma unroll
                for (int i = 0; i < 8; ++i)
                    atomicAdd(&normAcc[mbase + i], d[i]);
            }
        }
        __syncthreads();
        if (tid < TROWS)
            lognorm[tid] = __logf(normAcc[tid] + (float)CDIM * EPSV);
        __syncthreads();

        // ---- serial alpha recursion over this tile's rows ----
        const int rem  = T - tile * TROWS;
        const int rmax = (rem < TROWS) ? rem : TROWS;
        for (int r = 0; r < rmax; ++r) {
            const int t = tile * TROWS + r;
            if (tid < S) {
                float lp = __logf(P[r * ROWSTRIDE + extcls[tid]] + EPSV) - lognorm[r];
                if (t == 0) {
                    cur[tid + 2] = (tid < 2) ? lp : NEGV;
                } else {
                    float a1 = cur[tid + 2];                    // stay
                    float a2 = cur[tid + 1];                    // s-1
                    float a3 = allowsk[tid] ? cur[tid] : NEGV;  // s-2 (skip)
                    float mx = fmaxf(a1, fmaxf(a2, a3));
                    float l  = (mx <= NEGV * 0.5f) ? NEGV
                             : mx + __logf(__expf(a1 - mx) + __expf(a2 - mx) +
                                           __expf(a3 - mx));
                    nxt[tid + 2] = l + lp;
                }
            }
            __syncthreads();
            if (t > 0) { float* tmp = cur; cur = nxt; nxt = tmp; }
        }
    }

    // ---- loss = -lse2(alpha[S-1], alpha[S-2]) ----
    if (tid == 0) {
        float aN = cur[S + 1];
        float aM = cur[S];
        float m  = fmaxf(aN, aM);
        float l  = (m <= NEGV * 0.5f) ? NEGV
                 : m + __logf(__expf(aN - m) + __expf(aM - m));
        out[b] = -l;
    }
}

extern "C" void kernel_launch(void* const* d_in, const int* in_sizes, int n_in,
                              void* d_out, int out_size, void* d_ws, size_t ws_size,
                              hipStream_t stream) {
    (void)n_in; (void)d_ws; (void)ws_size;
    const int*   y_true = (const int*)d_in[0];
    const float* y_pred = (const float*)d_in[1];
    float*       out    = (float*)d_out;

    const int B = out_size;                    // output [B,1]
    const int L = in_sizes[0] / B;             // label length
    const int T = in_sizes[1] / (B * CDIM);    // time steps (C fixed at 256)

    ctc_fused_kernel<<<B, 256, 0, stream>>>(y_true, y_pred, out, T, L);
}